// GraphProp_15367392985683
// MI455X (gfx1250) — compile-verified
//
#include <hip/hip_runtime.h>
#include <hip/hip_bf16.h>

// Problem constants (match reference)
#define IN_DIM  128
#define OUT_DIM 64

typedef float v2f __attribute__((ext_vector_type(2)));
typedef float v8f __attribute__((ext_vector_type(8)));

// ---------------------------------------------------------------------------
// Zero d_out (harness poisons once; we must re-zero every call)
// ---------------------------------------------------------------------------
__global__ __launch_bounds__(256) void zero4_kernel(float* __restrict__ p, int n4) {
    int i = blockIdx.x * blockDim.x + threadIdx.x;
    if (i < n4) {
        float4 z = make_float4(0.f, 0.f, 0.f, 0.f);
        ((float4*)p)[i] = z;
    }
}

// ---------------------------------------------------------------------------
// GEMM: xw[n,64] = x[n,128] @ W[128,64] + b[64], via V_WMMA_F32_16X16X4_F32.
// One block = 16 node rows. 4 waves, one 16-wide column tile each.
// W (32KB) and the 16x128 x-tile (padded, 8.1KB) are staged in LDS.
//
// Register layouts per CDNA5 ISA (7.12.2, wave32):
//   A 16x4 f32 (2 VGPRs): lane%16 = M; lanes 0-15 -> K = k+{0,1},
//                                      lanes 16-31 -> K = k+{2,3}
//   B 4x16 f32 (2 VGPRs): lane%16 = N; same K striping as A.
//   C/D 16x16 f32 (8 VGPRs): VGPR r: lanes 0-15 -> M=r, lanes 16-31 -> M=r+8;
//                            N = lane%16.
// ---------------------------------------------------------------------------
#define XTILE_STRIDE 130  // pad 128 -> 130 floats to avoid 16-way LDS bank conflicts

__global__ __launch_bounds__(128)
void gemm_wmma_kernel(const float* __restrict__ x,
                      const float* __restrict__ W,
                      const float* __restrict__ b,
                      float* __restrict__ xw) {
    __shared__ float Ws[IN_DIM * OUT_DIM];      // 32 KB, whole weight matrix
    __shared__ float Xs[16 * XTILE_STRIDE];     // 16 rows of x, padded

    const int rowBase = blockIdx.x * 16;

    // Cooperative load: W (8192 floats) and 16x128 x tile (2048 floats)
    for (int i = threadIdx.x; i < IN_DIM * OUT_DIM; i += blockDim.x) {
        Ws[i] = W[i];
    }
    for (int i = threadIdx.x; i < 16 * IN_DIM; i += blockDim.x) {
        int r  = i >> 7;        // /128
        int kk = i & 127;
        Xs[r * XTILE_STRIDE + kk] = x[(size_t)(rowBase + r) * IN_DIM + kk];
    }
    __syncthreads();

    const int wave    = threadIdx.x >> 5;   // 0..3 -> column tile
    const int lane    = threadIdx.x & 31;
    const int half    = lane >> 4;          // 0 or 1
    const int lid     = lane & 15;          // M (for A) / N (for B,C,D)
    const int colBase = wave * 16;

    const float* xsrow = &Xs[lid * XTILE_STRIDE];

    v8f acc = {};
    #pragma unroll 4
    for (int k = 0; k < IN_DIM; k += 4) {
        const int k0 = k + half * 2;
        // A: two consecutive K values of this lane's row (8B-aligned in LDS)
        v2f a = *(const v2f*)(xsrow + k0);
        // B: W[k0][col], W[k0+1][col]
        v2f bb;
        bb.x = Ws[(k0)     * OUT_DIM + colBase + lid];
        bb.y = Ws[(k0 + 1) * OUT_DIM + colBase + lid];
        acc = __builtin_amdgcn_wmma_f32_16x16x4_f32(
            /*neg_a=*/false, a, /*neg_b=*/false, bb,
            /*c_mod=*/(short)0, acc, /*reuse_a=*/false, /*reuse_b=*/false);
    }

    const float bias = b[colBase + lid];
    #pragma unroll
    for (int r = 0; r < 8; ++r) {
        const int m = r + half * 8;
        xw[(size_t)(rowBase + m) * OUT_DIM + colBase + lid] = acc[r] + bias;
    }
}

// Scalar tail for node counts not divisible by 16 (0 blocks for N=100000).
__global__ __launch_bounds__(64)
void gemm_tail_kernel(const float* __restrict__ x,
                      const float* __restrict__ W,
                      const float* __restrict__ b,
                      float* __restrict__ xw,
                      int rowStart, int nRows) {
    int t   = blockIdx.x * blockDim.x + threadIdx.x;
    int row = rowStart + t / OUT_DIM;
    int col = t % OUT_DIM;
    if (row >= rowStart + nRows) return;
    float acc = b[col];
    const float* xr = x + (size_t)row * IN_DIM;
    for (int k = 0; k < IN_DIM; ++k) acc += xr[k] * W[k * OUT_DIM + col];
    xw[(size_t)row * OUT_DIM + col] = acc;
}

// ---------------------------------------------------------------------------
// SpMM scatter: out[row[e]] += val[e] * xw[col[e]]
// One thread = (edge, 4-dim quad): float4 gather + 4 f32 atomic adds.
// 16 consecutive threads stream one 256B row -> fully coalesced gathers.
// ---------------------------------------------------------------------------
__global__ __launch_bounds__(256)
void spmm_atomic_kernel(const float* __restrict__ xw,
                        const int*   __restrict__ erow,
                        const int*   __restrict__ ecol,
                        const float* __restrict__ eval,
                        float* __restrict__ out,
                        int n_edges) {
    long long t = (long long)blockIdx.x * blockDim.x + threadIdx.x;
    int e = (int)(t >> 4);
    int q = (int)(t & 15);
    if (e >= n_edges) return;

    const float v = eval[e];
    const int   c = ecol[e];
    const int   r = erow[e];

    const float4 m = *(const float4*)(xw + (size_t)c * OUT_DIM + q * 4);
    float* o = out + (size_t)r * OUT_DIM + q * 4;
    atomicAdd(o + 0, v * m.x);
    atomicAdd(o + 1, v * m.y);
    atomicAdd(o + 2, v * m.z);
    atomicAdd(o + 3, v * m.w);
}

// ---------------------------------------------------------------------------
// In-place ReLU on d_out
// ---------------------------------------------------------------------------
__global__ __launch_bounds__(256) void relu4_kernel(float* __restrict__ p, int n4) {
    int i = blockIdx.x * blockDim.x + threadIdx.x;
    if (i < n4) {
        float4 v = ((float4*)p)[i];
        v.x = fmaxf(v.x, 0.f);
        v.y = fmaxf(v.y, 0.f);
        v.z = fmaxf(v.z, 0.f);
        v.w = fmaxf(v.w, 0.f);
        ((float4*)p)[i] = v;
    }
}

// ---------------------------------------------------------------------------
// Launcher. Input order: x, edge_row, edge_col, edge_val, W, b
// ---------------------------------------------------------------------------
extern "C" void kernel_launch(void* const* d_in, const int* in_sizes, int n_in,
                              void* d_out, int out_size, void* d_ws, size_t ws_size,
                              hipStream_t stream) {
    const float* x    = (const float*)d_in[0];
    const int*   erow = (const int*)  d_in[1];
    const int*   ecol = (const int*)  d_in[2];
    const float* eval = (const float*)d_in[3];
    const float* W    = (const float*)d_in[4];
    const float* b    = (const float*)d_in[5];
    float*       out  = (float*)d_out;

    const int n_nodes = in_sizes[0] / IN_DIM;   // 100000
    const int n_edges = in_sizes[1];            // 1600000

    float* xw = (float*)d_ws;                   // [n_nodes, OUT_DIM] scratch

    // 1) zero the accumulator (d_out)
    const int total = n_nodes * OUT_DIM;        // multiple of 4
    {
        int n4 = total / 4;
        zero4_kernel<<<(n4 + 255) / 256, 256, 0, stream>>>(out, n4);
    }

    // 2) xw = x @ W + b  (WMMA f32 16x16x4)
    {
        const int fullTiles = n_nodes / 16;
        if (fullTiles > 0)
            gemm_wmma_kernel<<<fullTiles, 128, 0, stream>>>(x, W, b, xw);
        const int rem = n_nodes - fullTiles * 16;
        if (rem > 0) {
            int threads = rem * OUT_DIM;
            gemm_tail_kernel<<<(threads + 63) / 64, 64, 0, stream>>>(
                x, W, b, xw, fullTiles * 16, rem);
        }
    }

    // 3) edge scatter: out[row] += val * xw[col]
    {
        long long threads = (long long)n_edges * 16;
        int blocks = (int)((threads + 255) / 256);
        spmm_atomic_kernel<<<blocks, 256, 0, stream>>>(xw, erow, ecol, eval, out, n_edges);
    }

    // 4) ReLU in place
    {
        int n4 = total / 4;
        relu4_kernel<<<(n4 + 255) / 256, 256, 0, stream>>>(out, n4);
    }
}